// MultibandTransform_55216099557555
// MI455X (gfx1250) — compile-verified
//
#include <hip/hip_runtime.h>
#include <math.h>

// ---------------- problem constants ----------------
#define SRATE      44100
#define T_IN       441000
#define BATCH      8
#define NB         3
#define NCH        (NB * BATCH)          // 24 channels
#define FRAMES     1000                  // T_IN / 441
#define ORIG       441
#define NEWR       160
#define KW         475                   // 2*width + orig real taps
#define KWP        480                   // padded to multiple of 16
#define WIDTH      17
#define LDA        481                   // LDS stride (odd -> conflict-free column reads)
#define T_PAD      441088                // WIDTH + T_IN + tail zeros (>= 999*441+480)
#define NT_SCAN    1024
#define SEG        431                   // ceil(441000/1024); 1023*431=440913 < 441000

typedef __attribute__((ext_vector_type(2))) float v2f;
typedef __attribute__((ext_vector_type(8))) float v8f;

// ---------------- zero the pad regions of the staged signal ----------------
__global__ void zero_pads(float* __restrict__ sig) {
    float* base = sig + (long)blockIdx.x * T_PAD;
    for (int i = threadIdx.x; i < WIDTH; i += blockDim.x) base[i] = 0.0f;
    for (int i = WIDTH + T_IN + threadIdx.x; i < T_PAD; i += blockDim.x) base[i] = 0.0f;
}

// ---------------- sinc resample kernel table (fp64 build, matches torchaudio) ----------------
__global__ void build_sinc_kernel(float* __restrict__ Kmat) {
    int i = blockIdx.x * blockDim.x + threadIdx.x;
    if (i >= NEWR * KWP) return;
    int p = i / KWP, k = i % KWP;
    float v = 0.0f;
    if (k < KW) {
        const double base = 160.0 * 0.99;                 // min(441,160) * rolloff
        double t = -(double)p / 160.0 + ((double)k - (double)WIDTH) / 441.0;
        t *= base;
        t = fmin(fmax(t, -6.0), 6.0);
        double w = cos(t * M_PI / 12.0);                  // hann^2 window (LPF_WIDTH=6)
        w *= w;
        double tp = t * M_PI;
        double sinc = (tp == 0.0) ? 1.0 : sin(tp) / tp;
        v = (float)(sinc * w * (base / 441.0));
    }
    Kmat[i] = v;
}

// ---------------- block-split parallel biquad scan ----------------
// stage 0: bandpass  in = x[b], out = sig[ch] interior (clipped to [-1,1])
// stage 1: lowpass   in = sig[ch] * cos(2*pi*center*t/SR), out in-place (clipped)
__global__ __launch_bounds__(NT_SCAN) void iir_stage(const float* __restrict__ xglob,
                                                     float* __restrict__ sig,
                                                     int stage) {
    __shared__ float2 wst[NT_SCAN];    // zero-init segment end states
    __shared__ float2 inits[NT_SCAN];  // corrected segment initial states

    const int ch   = blockIdx.x;
    const int band = ch / BATCH;
    const int b    = ch % BATCH;

    // band geometry (centers are exact integers 4000/12000/19025)
    const double lo = (double)band * 8000.0;
    const double hi = fmin(lo + 8000.0, 22050.0);
    const double centerd = 0.5 * (lo + hi);
    const long long centerI = (long long)(centerd + 0.5);

    float b0, b1, b2, a1, a2;
    if (stage == 0) {
        double Q  = fmax(centerd / (hi - lo), 0.5);
        double w0 = 2.0 * M_PI * centerd / (double)SRATE;
        double al = sin(w0) / (2.0 * Q);
        double a0 = 1.0 + al;
        b0 = (float)(al / a0); b1 = 0.0f; b2 = (float)(-al / a0);
        a1 = (float)(-2.0 * cos(w0) / a0);
        a2 = (float)((1.0 - al) / a0);
    } else {
        double cut = 0.45 * 16000.0;
        double Q   = 0.7071067811865476;
        double w0  = 2.0 * M_PI * cut / (double)SRATE;
        double al  = sin(w0) / (2.0 * Q);
        double cw  = cos(w0);
        double a0  = 1.0 + al;
        b0 = (float)(((1.0 - cw) * 0.5) / a0);
        b1 = (float)((1.0 - cw) / a0);
        b2 = b0;
        a1 = (float)((-2.0 * cw) / a0);
        a2 = (float)((1.0 - al) / a0);
    }

    const float* in0 = (stage == 0) ? (xglob + (long)b * T_IN)
                                    : (sig + (long)ch * T_PAD + WIDTH);
    float* outp = sig + (long)ch * T_PAD + WIDTH;

    const int tid = threadIdx.x;
    const int s = tid * SEG;
    const int e = min(s + SEG, T_IN);

    auto mixv = [&](int t) -> float {
        float xv = in0[t];
        if (stage == 0) return xv;
        long long ph = (centerI * (long long)t) % (long long)SRATE;
        float arg = (float)ph * (float)(2.0 * M_PI / (double)SRATE);
        return xv * cosf(arg);
    };

    // halo FIR inputs captured once (safe for in-place stage 1)
    const float xh1 = (s >= 1) ? mixv(s - 1) : 0.0f;
    const float xh2 = (s >= 2) ? mixv(s - 2) : 0.0f;

    // --- phase 1: zero-initial-state segment run (particular solution) ---
    float x1 = xh1, x2 = xh2, y1 = 0.0f, y2 = 0.0f;
    for (int t = s; t < e; ++t) {
        float xt = mixv(t);
        float yt = fmaf(b0, xt, fmaf(b1, x1, fmaf(b2, x2, fmaf(-a1, y1, -a2 * y2))));
        x2 = x1; x1 = xt; y2 = y1; y1 = yt;
    }
    wst[tid] = make_float2(y1, y2);
    __syncthreads();

    // --- phase 2: stitch boundary states with M^SEG (thread 0, fp64) ---
    if (tid == 0) {
        double q00 = -(double)a1, q01 = -(double)a2, q10 = 1.0, q11 = 0.0;
        double p00 = 1.0, p01 = 0.0, p10 = 0.0, p11 = 1.0;
        int ee = SEG;
        while (ee) {
            if (ee & 1) {
                double r00 = p00 * q00 + p01 * q10, r01 = p00 * q01 + p01 * q11;
                double r10 = p10 * q00 + p11 * q10, r11 = p10 * q01 + p11 * q11;
                p00 = r00; p01 = r01; p10 = r10; p11 = r11;
            }
            double t00 = q00 * q00 + q01 * q10, t01 = q00 * q01 + q01 * q11;
            double t10 = q10 * q00 + q11 * q10, t11 = q10 * q01 + q11 * q11;
            q00 = t00; q01 = t01; q10 = t10; q11 = t11;
            ee >>= 1;
        }
        double s0 = 0.0, s1v = 0.0;    // [y_{-1}, y_{-2}]
        for (int i = 0; i < NT_SCAN; ++i) {
            inits[i] = make_float2((float)s0, (float)s1v);
            double n0 = p00 * s0 + p01 * s1v + (double)wst[i].x;
            double n1 = p10 * s0 + p11 * s1v + (double)wst[i].y;
            s0 = n0; s1v = n1;
        }
    }
    __syncthreads();

    // --- phase 3: re-run with corrected initial state, emit clipped output ---
    y1 = inits[tid].x; y2 = inits[tid].y;
    x1 = xh1; x2 = xh2;
    for (int t = s; t < e; ++t) {
        float xt = mixv(t);
        float yt = fmaf(b0, xt, fmaf(b1, x1, fmaf(b2, x2, fmaf(-a1, y1, -a2 * y2))));
        x2 = x1; x1 = xt; y2 = y1; y1 = yt;
        outp[t] = fminf(fmaxf(yt, -1.0f), 1.0f);
    }
}

// ---------------- polyphase resample as f32 WMMA GEMM ----------------
// Per wave: one 16(phases) x 16(frames) tile. K-loop over 480 padded taps,
// unrolled x4 with 4 independent accumulators so consecutive
// V_WMMA_F32_16X16X4_F32 ops have no C->C serial dependency.
// Signal is pre-padded in scratch -> all loads unconditional (EXEC all-1).
__global__ __launch_bounds__(256) void resample_wmma(const float* __restrict__ sig,
                                                     const float* __restrict__ Kmat,
                                                     float* __restrict__ out) {
    __shared__ float As[16 * LDA];

    const int blk   = blockIdx.x;
    const int ng    = blk & 7;          // 8 ntile-groups of 8 waves
    const int mtile = (blk >> 3) % 10;  // 160/16 phase tiles
    const int ch    = blk / 80;         // 24 channels
    const int band  = ch / BATCH;
    const int b     = ch % BATCH;
    const float* xin = sig + (long)ch * T_PAD;   // padded base (index 0 == x[-17])

    // stage A tile (16 phases x 480 taps) into LDS
    for (int i = threadIdx.x; i < 16 * KWP; i += 256) {
        int m = i / KWP, k = i % KWP;
        As[m * LDA + k] = Kmat[(mtile * 16 + m) * KWP + k];
    }
    __syncthreads();

    const int wave  = threadIdx.x >> 5;     // wave32
    const int lane  = threadIdx.x & 31;
    const int ntile = ng * 8 + wave;
    const int nidx  = lane & 15;            // A row (phase) / B col (frame-in-tile)
    const int half  = lane >> 4;            // K split within the wave
    const int frame = ntile * 16 + nidx;
    const bool fvalid = (frame < FRAMES);
    const int fload = fvalid ? frame : 0;   // invalid cols read frame 0, discarded at store

    const float* xb = xin + (long)fload * ORIG + half * 2;  // per-lane tap stream
    const int aoff = nidx * LDA + half * 2;

    __builtin_prefetch((const void*)xb, 0, 1);              // global_prefetch_b8

    v8f c0 = {}, c1 = {}, c2 = {}, c3 = {};
    for (int kk = 0; kk < KWP; kk += 16) {
        v2f a0v, a1v, a2v, a3v, b0v, b1v, b2v, b3v;
        a0v.x = As[aoff + kk +  0]; a0v.y = As[aoff + kk +  1];
        a1v.x = As[aoff + kk +  4]; a1v.y = As[aoff + kk +  5];
        a2v.x = As[aoff + kk +  8]; a2v.y = As[aoff + kk +  9];
        a3v.x = As[aoff + kk + 12]; a3v.y = As[aoff + kk + 13];
        b0v.x = xb[kk +  0]; b0v.y = xb[kk +  1];
        b1v.x = xb[kk +  4]; b1v.y = xb[kk +  5];
        b2v.x = xb[kk +  8]; b2v.y = xb[kk +  9];
        b3v.x = xb[kk + 12]; b3v.y = xb[kk + 13];
        c0 = __builtin_amdgcn_wmma_f32_16x16x4_f32(false, a0v, false, b0v, (short)0, c0, false, false);
        c1 = __builtin_amdgcn_wmma_f32_16x16x4_f32(false, a1v, false, b1v, (short)0, c1, false, false);
        c2 = __builtin_amdgcn_wmma_f32_16x16x4_f32(false, a2v, false, b2v, (short)0, c2, false, false);
        c3 = __builtin_amdgcn_wmma_f32_16x16x4_f32(false, a3v, false, b3v, (short)0, c3, false, false);
    }

    if (fvalid) {
        // C/D layout: VGPR r -> M = half*8 + r; column N = lane & 15
        float* op = out + ((long)b * NB + band) * ((long)FRAMES * NEWR)
                        + (long)frame * NEWR + mtile * 16 + half * 8;
#pragma unroll
        for (int r = 0; r < 8; ++r)
            op[r] = (c0[r] + c1[r]) + (c2[r] + c3[r]);
    }
}

// ---------------- launch ----------------
extern "C" void kernel_launch(void* const* d_in, const int* in_sizes, int n_in,
                              void* d_out, int out_size, void* d_ws, size_t ws_size,
                              hipStream_t stream) {
    const float* x = (const float*)d_in[0];
    float* out = (float*)d_out;

    float* sig  = (float*)d_ws;                       // 24 * T_PAD f32 (bp then lp in-place)
    float* Kmat = sig + (size_t)NCH * T_PAD;          // 160 * 480 f32, padded

    zero_pads<<<NCH, 256, 0, stream>>>(sig);
    build_sinc_kernel<<<(NEWR * KWP + 255) / 256, 256, 0, stream>>>(Kmat);
    iir_stage<<<NCH, NT_SCAN, 0, stream>>>(x, sig, 0);   // bandpass
    iir_stage<<<NCH, NT_SCAN, 0, stream>>>(x, sig, 1);   // heterodyne + lowpass
    resample_wmma<<<NCH * 10 * 8, 256, 0, stream>>>(sig, Kmat, out);
}